// DFTB_Layer_4922032521597
// MI455X (gfx1250) — compile-verified
//
#include <hip/hip_runtime.h>
#include <hip/hip_bf16.h>

// ---------------------------------------------------------------------------
// Problem constants (match reference)
// ---------------------------------------------------------------------------
#define NMOL 2048
#define BDIM 64
#define NROT 1000000
#define NTOT 4000000          // len(net_vals)
#define NX   200000
#define NREP 500000
#define NOCC 32

#define MM       (BDIM*BDIM)          // 4096
#define F_ELEMS  (NMOL*MM)            // 8388608

// d_out layout (floats), concatenated in reference return order:
//   Eelec[2048], Erep[2048], Eorb[2048*64], rho_out[2048*64*64], F[2048*64*64]
#define O_EELEC 0
#define O_EREP  (NMOL)
#define O_EORB  (2*NMOL)
#define O_RHO   (2*NMOL + NMOL*BDIM)
#define O_F     (O_RHO + F_ELEMS)

// workspace layout (floats)
#define W_NET 0
#define W_ROT (NTOT)                          // 2 + 3*NROT values (padded)
#define W_EP  (NTOT + 3*NROT + 64)            // ep[NMOL*64]
#define W_E2  (W_EP + NMOL*BDIM)              // ener2[NMOL]

typedef __attribute__((ext_vector_type(2))) float v2f;
typedef __attribute__((ext_vector_type(8))) float v8f;

#define LD 65          // padded LDS row stride (floats) -> conflict-free rows & cols
#define NSWEEP 10      // Jacobi sweeps

// ---------------------------------------------------------------------------
// f32 WMMA: D(16x16) += A(16x4) * B(4x16), f32 in / f32 out
// ---------------------------------------------------------------------------
__device__ inline v8f wmma4(v2f a, v2f b, v8f c) {
  return __builtin_amdgcn_wmma_f32_16x16x4_f32(
      /*neg_a=*/false, a, /*neg_b=*/false, b,
      /*c_mod=*/(short)0, c, /*reuse_a=*/false, /*reuse_b=*/false);
}

// D(64x64) = scale * op(A) * op(B); operands live in LDS with row stride LD.
// One block = 4 waves; wave w owns tiles t = w, w+4, w+8, w+12 (tn fixed per
// wave -> B operands are loop-invariant and get hoisted by the compiler).
// Loop control is forced scalar via readfirstlane so EXEC stays all-1s
// (WMMA ISA requirement) and no per-lane exec masking is generated.
// bColMap (optional, transB==0 only): B column permutation B[k][map[n]].
__device__ inline void gemm64(const float* sA, int transA,
                              const float* sB, int transB, const int* bColMap,
                              float* sD, int K, float scale, int tid) {
  const int wave = __builtin_amdgcn_readfirstlane(tid >> 5);
  const int lane = tid & 31;
  const int hi = lane >> 4;        // half-wave selector (K split / M split)
  const int l  = lane & 15;
  for (int t = wave; t < 16; t += 4) {
    const int tm = (t >> 2) << 4;
    const int tn = (t & 3) << 4;
    v8f acc = {};
    for (int k0 = 0; k0 < K; k0 += 4) {
      const int ka = k0 + 2*hi;    // lanes 0-15: K=k0,k0+1 ; lanes 16-31: K=k0+2,k0+3
      v2f a, b;
      if (transA) {
        a.x = sA[ka*LD + tm + l];
        a.y = sA[(ka+1)*LD + tm + l];
      } else {
        a.x = sA[(tm + l)*LD + ka];
        a.y = sA[(tm + l)*LD + ka + 1];
      }
      if (transB) {
        b.x = sB[(tn + l)*LD + ka];
        b.y = sB[(tn + l)*LD + ka + 1];
      } else {
        const int col = bColMap ? bColMap[tn + l] : (tn + l);
        b.x = sB[ka*LD + col];
        b.y = sB[(ka+1)*LD + col];
      }
      acc = wmma4(a, b, acc);
    }
#pragma unroll
    for (int r = 0; r < 8; ++r)    // C/D layout: VGPR r -> row r (lanes<16) / r+8
      sD[(tm + r + 8*hi)*LD + tn + l] = scale * acc[r];
  }
}

// ---------------------------------------------------------------------------
// Stage 1: net_vals = net_base; net_vals[scatter_idx] = x
// ---------------------------------------------------------------------------
__global__ void k_net_copy(const float* __restrict__ base, float* __restrict__ net) {
  int i = blockIdx.x*blockDim.x + threadIdx.x;
  if (i < NTOT) net[i] = base[i];
}
__global__ void k_net_scatter(const float* __restrict__ x, const int* __restrict__ idx,
                              float* __restrict__ net) {
  int i = blockIdx.x*blockDim.x + threadIdx.x;
  if (i < NX) net[idx[i]] = x[i];
}

// ---------------------------------------------------------------------------
// Stage 2: rot_out = [0, 1, (rot_tensors @ net_vals[rot_gather]).flatten()]
// ---------------------------------------------------------------------------
__global__ void k_rot(const float* __restrict__ rt, const int* __restrict__ rg,
                      const float* __restrict__ net, float* __restrict__ rot) {
  int r = blockIdx.x*blockDim.x + threadIdx.x;
  if (r == 0) { rot[0] = 0.0f; rot[1] = 1.0f; }
  if (r < NROT) {
    float v0 = net[rg[3*r+0]];
    float v1 = net[rg[3*r+1]];
    float v2 = net[rg[3*r+2]];
    const float* t = rt + 9*r;
    rot[2 + 3*r + 0] = t[0]*v0 + t[1]*v1 + t[2]*v2;
    rot[2 + 3*r + 1] = t[3]*v0 + t[4]*v1 + t[5]*v2;
    rot[2 + 3*r + 2] = t[6]*v0 + t[7]*v1 + t[8]*v2;
  }
}

// ---------------------------------------------------------------------------
// Stage 3: H = rot_out[oper_gather]  (written directly into the F output slot)
// ---------------------------------------------------------------------------
__global__ void k_oper(const float* __restrict__ rot, const int* __restrict__ og,
                       float* __restrict__ H) {
  int i = blockIdx.x*blockDim.x + threadIdx.x;
  if (i < F_ELEMS) H[i] = rot[og[i]];
}

// ---------------------------------------------------------------------------
// Stage 4: Fock assembly, in place: F = H + couMat. Stores ep and ener2.
// One 64-thread block per molecule. All global traffic is coalesced
// (consecutive lanes -> consecutive addresses); row sums read from a padded
// LDS tile (conflict-free).
// ---------------------------------------------------------------------------
__global__ __launch_bounds__(64)
void k_fock(const float* __restrict__ S, const float* __restrict__ G,
            const float* __restrict__ rho, const float* __restrict__ qn,
            float* __restrict__ F, float* __restrict__ ep_ws, float* __restrict__ e2_ws) {
  __shared__ float sTile[BDIM*LD];
  __shared__ float sdQ[BDIM], sep[BDIM], sred[BDIM];
  const int m = blockIdx.x, i = threadIdx.x;
  const float* Sm = S   + m*MM;
  const float* Gm = G   + m*MM;
  const float* Rm = rho + m*MM;

  // qbasis = rho_in * S, staged to LDS (coalesced: lane = column)
#pragma unroll 4
  for (int it = 0; it < BDIM; ++it) {
    const int idx = it*BDIM + i;               // row=it, col=i
    sTile[it*LD + i] = Rm[idx] * Sm[idx];
  }
  __syncthreads();
  float gop = 0.0f;
  for (int j = 0; j < BDIM; ++j) gop += sTile[i*LD + j];
  const float dq = qn[m*BDIM + i] - gop;
  sdQ[i] = dq;
  __syncthreads();

  // stage G, then ep = G @ dQ
#pragma unroll 4
  for (int it = 0; it < BDIM; ++it)
    sTile[it*LD + i] = Gm[it*BDIM + i];
  __syncthreads();
  float e = 0.0f;
  for (int j = 0; j < BDIM; ++j) e += sTile[i*LD + j] * sdQ[j];
  sep[i] = e;
  ep_ws[m*BDIM + i] = e;
  sred[i] = dq * e;
  __syncthreads();
  for (int s = 32; s > 0; s >>= 1) {
    if (i < s) sred[i] += sred[i + s];
    __syncthreads();
  }
  if (i == 0) e2_ws[m] = 0.5f * sred[0];
  __syncthreads();

  // F = H - 0.5*S*(ep_r + ep_c), coalesced in-place update
#pragma unroll 4
  for (int it = 0; it < BDIM; ++it) {
    const int idx = it*BDIM + i;               // row=it, col=i
    F[m*MM + idx] -= 0.5f * Sm[idx] * (sep[it] + sep[i]);
  }
}

// ---------------------------------------------------------------------------
// Stage 5: Erep = segment_sum(net_vals[rep_gather], rep_seg)
// ---------------------------------------------------------------------------
__global__ void k_erep_zero(float* __restrict__ er) {
  int i = blockIdx.x*blockDim.x + threadIdx.x;
  if (i < NMOL) er[i] = 0.0f;
}
__global__ void k_erep(const float* __restrict__ net, const int* __restrict__ rg,
                       const int* __restrict__ seg, float* __restrict__ er) {
  int i = blockIdx.x*blockDim.x + threadIdx.x;
  if (i < NREP) atomicAdd(&er[seg[i]], net[rg[i]]);
}

// ---------------------------------------------------------------------------
// Stage 6 (mega kernel, one 128-thread / 4-wave block per molecule):
//   fockp = phiS^T @ F @ phiS        (WMMA f32 16x16x4, in LDS)
//   Eorb, V = jacobi_eigh(fockp)     (parallel cyclic Jacobi in LDS)
//   orb   = phiS @ V[:, sort-perm]   (WMMA, column-remapped B)
//   rho   = 2 * orb[:, :32] @ orb[:, :32]^T   (WMMA)
//   ener1 = sum(rho * H),  H rebuilt as F + 0.5*S*(ep_i + ep_j)
//   Eelec = ener1 + ener2
// ---------------------------------------------------------------------------
__global__ __launch_bounds__(128)
void k_mega(const float* __restrict__ Fg, const float* __restrict__ phiS,
            const float* __restrict__ Sg, const float* __restrict__ ep_ws,
            const float* __restrict__ e2_ws, float* __restrict__ Eelec,
            float* __restrict__ Eorb, float* __restrict__ rho_out) {
  __shared__ float sA[BDIM*LD];   // F -> fockp/A -> orb
  __shared__ float sP[BDIM*LD];   // phiS
  __shared__ float sT[BDIM*LD];   // T -> V -> rho
  __shared__ float sE[BDIM];      // eigenvalues, then ep
  __shared__ float sCS[64];       // (c,s) per rotation pair
  __shared__ int   sPi[32], sQi[32], sSrc[BDIM];
  __shared__ float sRed[128];

  const int m = blockIdx.x, tid = threadIdx.x;
  const float* Fm = Fg   + m*MM;
  const float* Pm = phiS + m*MM;
  const float* Sm = Sg   + m*MM;

  // Warm L2 for the S tile consumed after the (long) Jacobi phase.
  __builtin_prefetch(Sm + tid*32, 0, 1);       // 128 lanes x 128B = whole tile

#pragma unroll
  for (int it = 0; it < MM/128; ++it) {
    const int idx = it*128 + tid;
    const int r = idx >> 6, c = idx & 63;
    sA[r*LD + c] = Fm[idx];
    sP[r*LD + c] = Pm[idx];
  }
  __syncthreads();

  gemm64(sA, 0, sP, 0, nullptr, sT, BDIM, 1.0f, tid);     // T = F @ phiS
  __syncthreads();
  gemm64(sP, 1, sT, 0, nullptr, sA, BDIM, 1.0f, tid);     // A = phiS^T @ T
  __syncthreads();

#pragma unroll
  for (int it = 0; it < MM/128; ++it) {                   // V = I
    const int idx = it*128 + tid;
    const int r = idx >> 6, c = idx & 63;
    sT[r*LD + c] = (r == c) ? 1.0f : 0.0f;
  }
  __syncthreads();

  // -------- parallel cyclic Jacobi: 63 rounds of 32 disjoint pairs --------
  for (int sweep = 0; sweep < NSWEEP; ++sweep) {
    for (int round = 0; round < 63; ++round) {
      if (tid < 32) {
        int p, q;
        if (tid == 0) { p = 63; q = round; }
        else { p = (round + tid) % 63; q = (round - tid + 63) % 63; }
        const float app = sA[p*LD + p], aqq = sA[q*LD + q], apq = sA[p*LD + q];
        float c_ = 1.0f, s_ = 0.0f;
        if (fabsf(apq) > 1e-18f) {
          const float th = (aqq - app) / (2.0f * apq);
          float t = 1.0f / (fabsf(th) + sqrtf(1.0f + th*th));
          t = (th < 0.0f) ? -t : t;
          c_ = rsqrtf(1.0f + t*t);
          s_ = t * c_;
        }
        sCS[2*tid] = c_; sCS[2*tid + 1] = s_;
        sPi[tid] = p; sQi[tid] = q;
      }
      __syncthreads();
#pragma unroll 2
      for (int it = 0; it < 16; ++it) {                   // A <- J^T A (rows)
        const int w = it*128 + tid;
        const int pr = w >> 6, k = w & 63;
        const int p = sPi[pr], q = sQi[pr];
        const float c_ = sCS[2*pr], s_ = sCS[2*pr + 1];
        const float ap = sA[p*LD + k], aq = sA[q*LD + k];
        sA[p*LD + k] = c_*ap - s_*aq;
        sA[q*LD + k] = s_*ap + c_*aq;
      }
      __syncthreads();
#pragma unroll 2
      for (int it = 0; it < 16; ++it) {                   // A <- A J, V <- V J (cols)
        const int w = it*128 + tid;
        const int pr = w >> 6, k = w & 63;
        const int p = sPi[pr], q = sQi[pr];
        const float c_ = sCS[2*pr], s_ = sCS[2*pr + 1];
        const float ap = sA[k*LD + p], aq = sA[k*LD + q];
        sA[k*LD + p] = c_*ap - s_*aq;
        sA[k*LD + q] = s_*ap + c_*aq;
        const float vp = sT[k*LD + p], vq = sT[k*LD + q];
        sT[k*LD + p] = c_*vp - s_*vq;
        sT[k*LD + q] = s_*vp + c_*vq;
      }
      __syncthreads();
    }
  }

  // -------- sort eigenvalues ascending; build source-column permutation ----
  if (tid < BDIM) sE[tid] = sA[tid*LD + tid];
  __syncthreads();
  if (tid < BDIM) {
    const float my = sE[tid];
    int rank = 0;
    for (int j = 0; j < BDIM; ++j) {
      const float o = sE[j];
      rank += (o < my) || (o == my && j < tid);
    }
    Eorb[m*BDIM + rank] = my;
    sSrc[rank] = tid;            // vecs[:, rank] = V[:, tid]
  }
  __syncthreads();

  gemm64(sP, 0, sT, 0, sSrc, sA, BDIM, 1.0f, tid);        // orb = phiS @ vecs
  __syncthreads();
  gemm64(sA, 0, sA, 1, nullptr, sT, NOCC, 2.0f, tid);     // rho = 2*orb_occ@orb_occ^T
  __syncthreads();

  if (tid < BDIM) sE[tid] = ep_ws[m*BDIM + tid];          // reuse sE for ep
  __syncthreads();

  float part = 0.0f;
#pragma unroll
  for (int it = 0; it < MM/128; ++it) {
    const int idx = it*128 + tid;
    const int r = idx >> 6, c = idx & 63;
    const float rho = sT[r*LD + c];
    rho_out[m*MM + idx] = rho;
    const float h = Fm[idx] + 0.5f * Sm[idx] * (sE[r] + sE[c]);  // H = F - couMat
    part += rho * h;
  }
  sRed[tid] = part;
  __syncthreads();
  for (int s = 64; s > 0; s >>= 1) {
    if (tid < s) sRed[tid] += sRed[tid + s];
    __syncthreads();
  }
  if (tid == 0) Eelec[m] = sRed[0] + e2_ws[m];
}

// ---------------------------------------------------------------------------
// Launcher
// ---------------------------------------------------------------------------
extern "C" void kernel_launch(void* const* d_in, const int* in_sizes, int n_in,
                              void* d_out, int out_size, void* d_ws, size_t ws_size,
                              hipStream_t stream) {
  (void)in_sizes; (void)n_in; (void)out_size; (void)ws_size;
  const float* x        = (const float*)d_in[0];
  const float* net_base = (const float*)d_in[1];
  const float* rot_t    = (const float*)d_in[2];
  const float* S        = (const float*)d_in[3];
  const float* G        = (const float*)d_in[4];
  const float* rho_in   = (const float*)d_in[5];
  const float* phiS     = (const float*)d_in[6];
  const float* qn       = (const float*)d_in[7];
  /* d_in[8] occ_mask: equivalent to (col < NOCC), hardcoded */
  const int* scatter_idx = (const int*)d_in[9];
  const int* rot_gather  = (const int*)d_in[10];
  const int* oper_gather = (const int*)d_in[11];
  const int* rep_gather  = (const int*)d_in[12];
  const int* rep_seg     = (const int*)d_in[13];

  float* out = (float*)d_out;
  float* ws  = (float*)d_ws;
  float* net = ws + W_NET;
  float* rot = ws + W_ROT;
  float* ep  = ws + W_EP;
  float* e2  = ws + W_E2;

  k_net_copy   <<<(NTOT + 255)/256, 256, 0, stream>>>(net_base, net);
  k_net_scatter<<<(NX   + 255)/256, 256, 0, stream>>>(x, scatter_idx, net);
  k_rot        <<<(NROT + 255)/256, 256, 0, stream>>>(rot_t, rot_gather, net, rot);
  k_oper       <<<(F_ELEMS + 255)/256, 256, 0, stream>>>(rot, oper_gather, out + O_F);
  k_fock       <<<NMOL, 64, 0, stream>>>(S, G, rho_in, qn, out + O_F, ep, e2);
  k_erep_zero  <<<(NMOL + 255)/256, 256, 0, stream>>>(out + O_EREP);
  k_erep       <<<(NREP + 255)/256, 256, 0, stream>>>(net, rep_gather, rep_seg, out + O_EREP);
  k_mega       <<<NMOL, 128, 0, stream>>>(out + O_F, phiS, S, ep, e2,
                                          out + O_EELEC, out + O_EORB, out + O_RHO);
}